// SimpleHeadAttentionMultiLatent_5085241278575
// MI455X (gfx1250) — compile-verified
//
#include <hip/hip_runtime.h>
#include <hip/hip_bf16.h>

// ---------------------------------------------------------------------------
// MLA attention forward for MI455X (gfx1250), fp32 path via V_WMMA_F32_16X16X4_F32.
// Problem: B=8, S=1024, D=512, H=8, HD=64, L=128, LH=16.
// Outputs (concatenated flat in d_out): out (B,S,D), attn_weights (B,H,S,S),
// qk_values (B,H,S,S).
// ---------------------------------------------------------------------------

typedef __attribute__((ext_vector_type(2))) float v2f;
typedef __attribute__((ext_vector_type(8))) float v8f;

#define CB 8
#define CS 1024
#define CD 512
#define CH 8
#define CHD 64
#define CL 128
#define CLH 16

// ---------------------------------------------------------------------------
// Generic fp32 WMMA GEMM: C[z] = alpha * A[z] * B[z] + bias
//   A: (M,K) row-major, lda
//   B: BT==false -> (K,N) row-major, ldb ; BT==true -> Bt is (N,K) row-major, ldb
//   C: (M,N) row-major, ldc
//
// Each wave owns a 16x64 strip of C (4 adjacent 16x16 N-tiles) so the A
// fragment is loaded once and reused by 4 WMMAs per K-step (4 wmma : 5 b64).
// 4 waves per block stripe over adjacent M-tiles.
//   grid.x = N/64, grid.y = M/64, blockDim = (32,4)
// Requires M % 64 == 0 and N % 64 == 0 (true for all launches here).
//
// fp32 WMMA operand layout (ISA 7.12.2):
//   A 16x4 : lanes 0-15 M=lane, VGPR0=K0, VGPR1=K1 ; lanes 16-31 VGPR0=K2, VGPR1=K3
//   B 4x16 : lanes 0-15 N=lane, VGPR0=K0, VGPR1=K1 ; lanes 16-31 VGPR0=K2, VGPR1=K3
//   C 16x16: VGPR i holds row i (lanes 0-15, col=lane) / row i+8 (lanes 16-31)
// ---------------------------------------------------------------------------
template <bool BT>
__global__ __launch_bounds__(128) void gemm_wmma_f32(
    const float* __restrict__ A, const float* __restrict__ B,
    float* __restrict__ C, const float* __restrict__ bias,
    int M, int N, int K, int lda, int ldb, int ldc, float alpha, int z2c,
    long sA1, long sA0, long sB1, long sB0, long sC1, long sC0)
{
    const int z  = blockIdx.z;
    const int z1 = z / z2c;
    const int z0 = z % z2c;
    A += (long)z1 * sA1 + (long)z0 * sA0;
    B += (long)z1 * sB1 + (long)z0 * sB0;
    C += (long)z1 * sC1 + (long)z0 * sC0;

    const int lane = threadIdx.x;                       // 0..31 (wave32)
    const int tm   = blockIdx.y * blockDim.y + threadIdx.y;  // 16-row M tile
    const int tn0  = blockIdx.x * 4;                    // first of 4 N tiles
    if (tm * 16 >= M || tn0 * 16 >= N) return;          // wave-uniform

    const int hlane = lane & 15;
    const int ksel  = (lane >> 4) * 2;                  // 0 for lanes 0-15, 2 for 16-31
    const int arow  = tm * 16 + hlane;                  // A row held by this lane
    const int bcol0 = tn0 * 16 + hlane;                 // first B col held by this lane

    v8f c0 = {}, c1 = {}, c2 = {}, c3 = {};
    for (int k0 = 0; k0 < K; k0 += 4) {
        v2f a, b0, b1, b2, b3;
        const float* ap = A + (long)arow * lda + (k0 + ksel);
        a.x = ap[0];
        a.y = ap[1];
        if (BT) {
            const float* bp = B + (long)bcol0 * ldb + (k0 + ksel);
            b0.x = bp[0];            b0.y = bp[1];
            b1.x = bp[16 * ldb];     b1.y = bp[16 * ldb + 1];
            b2.x = bp[32 * ldb];     b2.y = bp[32 * ldb + 1];
            b3.x = bp[48 * ldb];     b3.y = bp[48 * ldb + 1];
        } else {
            const float* bp = B + (long)(k0 + ksel) * ldb + bcol0;
            b0.x = bp[0];   b0.y = bp[ldb];
            b1.x = bp[16];  b1.y = bp[ldb + 16];
            b2.x = bp[32];  b2.y = bp[ldb + 32];
            b3.x = bp[48];  b3.y = bp[ldb + 48];
        }
        c0 = __builtin_amdgcn_wmma_f32_16x16x4_f32(false, a, false, b0, (short)0, c0, false, false);
        c1 = __builtin_amdgcn_wmma_f32_16x16x4_f32(false, a, false, b1, (short)0, c1, false, false);
        c2 = __builtin_amdgcn_wmma_f32_16x16x4_f32(false, a, false, b2, (short)0, c2, false, false);
        c3 = __builtin_amdgcn_wmma_f32_16x16x4_f32(false, a, false, b3, (short)0, c3, false, false);
    }

    const int orow = tm * 16 + (lane >> 4) * 8;
    const int ocol = tn0 * 16 + hlane;
    v8f acc[4] = {c0, c1, c2, c3};
#pragma unroll
    for (int nt = 0; nt < 4; ++nt) {
        const int cc = ocol + nt * 16;
        const float bv = bias ? bias[cc] : 0.0f;
#pragma unroll
        for (int i = 0; i < 8; ++i) {
            C[(long)(orow + i) * ldc + cc] = acc[nt][i] * alpha + bv;
        }
    }
}

// ---------------------------------------------------------------------------
// b_abs[l] = sum_k bq[k] * Wuk[l,k]   (tiny: 128 x 512)
// ---------------------------------------------------------------------------
__global__ void absorb_bias_kernel(const float* __restrict__ bq,
                                   const float* __restrict__ Wuk,
                                   float* __restrict__ b_abs)
{
    const int l = threadIdx.x;                          // 0..127
    float s = 0.0f;
    for (int k = 0; k < CD; ++k) s += bq[k] * Wuk[(long)l * CD + k];
    b_abs[l] = s;
}

// ---------------------------------------------------------------------------
// Masked stable softmax over rows of length S.
// blockIdx.x = (b*H + h)*S + i ; 256 threads, 4 elements each.
// logits[j] = qk[row, j] + min(1, mask[b,i] + mask[b,j]) * (-1e9)
// ---------------------------------------------------------------------------
__device__ inline float wave_max(float v) {
#pragma unroll
    for (int o = 16; o > 0; o >>= 1) v = fmaxf(v, __shfl_xor(v, o, 32));
    return v;
}
__device__ inline float wave_sum(float v) {
#pragma unroll
    for (int o = 16; o > 0; o >>= 1) v += __shfl_xor(v, o, 32);
    return v;
}

__global__ __launch_bounds__(256) void softmax_mask_kernel(
    const float* __restrict__ qk, const float* __restrict__ mask,
    float* __restrict__ attn_w)
{
    __shared__ float sred[8];
    const int gid = blockIdx.x;             // (b*H+h)*S + i
    const int i   = gid & (CS - 1);
    const int bh  = gid >> 10;              // S == 1024
    const int b   = bh >> 3;                // H == 8
    const float mi = mask[(long)b * CS + i];

    const float* row = qk + (long)gid * CS;
    float vals[4];
    float vmax = -3.4e38f;
#pragma unroll
    for (int t = 0; t < 4; ++t) {
        const int j = threadIdx.x + t * 256;
        float m = fminf(1.0f, mi + mask[(long)b * CS + j]);
        float v = row[j] + m * (-1e9f);
        vals[t] = v;
        vmax = fmaxf(vmax, v);
    }

    const int wid = threadIdx.x >> 5;
    const int ln  = threadIdx.x & 31;

    vmax = wave_max(vmax);
    if (ln == 0) sred[wid] = vmax;
    __syncthreads();
    if (wid == 0) {
        float t = (ln < 8) ? sred[ln] : -3.4e38f;
        t = wave_max(t);
        if (ln == 0) sred[0] = t;
    }
    __syncthreads();
    vmax = sred[0];
    __syncthreads();

    float s = 0.0f;
#pragma unroll
    for (int t = 0; t < 4; ++t) {
        vals[t] = expf(vals[t] - vmax);
        s += vals[t];
    }
    s = wave_sum(s);
    if (ln == 0) sred[wid] = s;
    __syncthreads();
    if (wid == 0) {
        float t = (ln < 8) ? sred[ln] : 0.0f;
        t = wave_sum(t);
        if (ln == 0) sred[0] = t;
    }
    __syncthreads();
    const float inv = 1.0f / sred[0];

    float* orow = attn_w + (long)gid * CS;
#pragma unroll
    for (int t = 0; t < 4; ++t) orow[threadIdx.x + t * 256] = vals[t] * inv;
}

// ---------------------------------------------------------------------------
extern "C" void kernel_launch(void* const* d_in, const int* in_sizes, int n_in,
                              void* d_out, int out_size, void* d_ws, size_t ws_size,
                              hipStream_t stream)
{
    (void)in_sizes; (void)n_in; (void)out_size; (void)ws_size;

    const float* x    = (const float*)d_in[0];   // (B,S,D)
    const float* mask = (const float*)d_in[1];   // (B,S,1)
    const float* Wq   = (const float*)d_in[2];   // (D,D)
    const float* bq   = (const float*)d_in[3];   // (D)
    const float* Wckv = (const float*)d_in[4];   // (D,L)
    const float* Wuk  = (const float*)d_in[5];   // (L,D)
    const float* Wuv  = (const float*)d_in[6];   // (L,D)
    const float* Wo   = (const float*)d_in[7];   // (D,D)
    const float* bo   = (const float*)d_in[8];   // (D)

    float* out = (float*)d_out;
    float* attn_w = out + (long)CB * CS * CD;                       // 4,194,304
    float* qk_out = attn_w + (long)CB * CH * CS * CS;               // +67,108,864

    float* ws      = (float*)d_ws;
    float* W_qabs  = ws;                                  // 512*128   = 65,536
    float* b_abs   = W_qabs + (long)CD * CL;              // 128 (pad 512)
    float* q_lat   = b_abs + 512;                         // 8192*128  = 1,048,576
    float* k_lat   = q_lat + (long)CB * CS * CL;
    float* v_full  = k_lat + (long)CB * CS * CL;          // 8192*512  = 4,194,304
    float* attn_ws = v_full + (long)CB * CS * CD;         // 8192*512

    const dim3 blk(32, 4);
    const long S2 = (long)CS * CS;

    // 1) W_qabs = Wq @ Wuk^T     (M=512, N=128, K=512), B transposed (Bt = Wuk)
    gemm_wmma_f32<true><<<dim3(CL / 64, CD / 64, 1), blk, 0, stream>>>(
        Wq, Wuk, W_qabs, nullptr, CD, CL, CD, CD, CD, CL, 1.0f, 1,
        0, 0, 0, 0, 0, 0);

    // 1b) b_abs = bq @ Wuk^T
    absorb_bias_kernel<<<1, CL, 0, stream>>>(bq, Wuk, b_abs);

    // 2) q_lat = x @ W_qabs + b_abs   (M=8192, N=128, K=512)
    gemm_wmma_f32<false><<<dim3(CL / 64, (CB * CS) / 64, 1), blk, 0, stream>>>(
        x, W_qabs, q_lat, b_abs, CB * CS, CL, CD, CD, CL, CL, 1.0f, 1,
        0, 0, 0, 0, 0, 0);

    // 3) k_lat = x @ Wckv             (M=8192, N=128, K=512)
    gemm_wmma_f32<false><<<dim3(CL / 64, (CB * CS) / 64, 1), blk, 0, stream>>>(
        x, Wckv, k_lat, nullptr, CB * CS, CL, CD, CD, CL, CL, 1.0f, 1,
        0, 0, 0, 0, 0, 0);

    // 4) v_full = k_lat @ Wuv         (M=8192, N=512, K=128)
    gemm_wmma_f32<false><<<dim3(CD / 64, (CB * CS) / 64, 1), blk, 0, stream>>>(
        k_lat, Wuv, v_full, nullptr, CB * CS, CD, CL, CL, CD, CD, 1.0f, 1,
        0, 0, 0, 0, 0, 0);

    // 5) qk_values[b,h] = q_lat[b,:,h*16:+16] @ k_lat[b,:,h*16:+16]^T / 4
    //    batched over z = b*H + h ; M=N=1024, K=16
    gemm_wmma_f32<true><<<dim3(CS / 64, CS / 64, CB * CH), blk, 0, stream>>>(
        q_lat, k_lat, qk_out, nullptr, CS, CS, CLH, CL, CL, CS, 0.25f, CH,
        (long)CS * CL, CLH, (long)CS * CL, CLH, (long)CH * S2, S2);

    // 6) attn_weights = softmax(qk + mask*(-1e9))
    softmax_mask_kernel<<<CB * CH * CS, 256, 0, stream>>>(qk_out, mask, attn_w);

    // 7) attn[b,:,h*64:+64] = attn_weights[b,h] @ v_full[b,:,h*64:+64]
    //    M=1024, N=64, K=1024
    gemm_wmma_f32<false><<<dim3(CHD / 64, CS / 64, CB * CH), blk, 0, stream>>>(
        attn_w, v_full, attn_ws, nullptr, CS, CHD, CS, CS, CD, CD, 1.0f, CH,
        (long)CH * S2, S2, (long)CS * CD, CHD, (long)CS * CD, CHD);

    // 8) out = attn @ Wo + bo         (M=8192, N=512, K=512)
    gemm_wmma_f32<false><<<dim3(CD / 64, (CB * CS) / 64, 1), blk, 0, stream>>>(
        attn_ws, Wo, out, bo, CB * CS, CD, CD, CD, CD, CD, 1.0f, 1,
        0, 0, 0, 0, 0, 0);
}